// VectorQuantize_84748294685202
// MI455X (gfx1250) — compile-verified
//
#include <hip/hip_runtime.h>
#include <hip/hip_bf16.h>

typedef __attribute__((ext_vector_type(2))) float v2f;
typedef __attribute__((ext_vector_type(4))) float v4f;
typedef __attribute__((ext_vector_type(8))) float v8f;

#define N_EMBED 1024
#define EDIM    64
#define NPIX    65536          // 16 * 64 * 64
#define NELEM   4194304        // 16 * 64 * 64 * 64

// d_out layout (floats): [q: 0 .. 4194303][loss: 4194304][perp: 4194305]
// [onehot: 4194306 .. +67108863][index: 71303170 .. +65535]
#define OFF_LOSS   4194304
#define OFF_PERP   4194305
#define OFF_ONEHOT 4194306
#define OFF_INDEX  71303170

// ---------------------------------------------------------------------------
// k0: zero workspace (1024 histogram bins + 1 loss accumulator)
// ---------------------------------------------------------------------------
__global__ void vq_zero_ws_kernel(float* __restrict__ ws) {
    int t = blockIdx.x * 256 + threadIdx.x;
    if (t < 1025) ws[t] = 0.0f;
}

// ---------------------------------------------------------------------------
// k1: WMMA f32 distance + argmin.
//  grid: 256 blocks x 256 threads (8 waves). Each wave owns 2 pixel-tiles of
//  16 consecutive pixels (same b,h; 16 consecutive w). Codebook staged in LDS
//  in 8 chunks of 128 codes, row stride 66 floats (conflict-free).
//  ct loop kept rolled (unroll 1) to hold only 2 live accumulator tiles and
//  stay under 256 VGPRs (avoids s_set_vgpr_msb churn / occupancy collapse).
// ---------------------------------------------------------------------------
__global__ __launch_bounds__(256) void vq_argmin_kernel(
    const float* __restrict__ z, const float* __restrict__ ew,
    float* __restrict__ out_index, float* __restrict__ ws_counts)
{
    __shared__ float ebuf[128 * 66];   // padded rows: conflict-free staging
    __shared__ float e2[N_EMBED];      // ||e||^2 per code

    const int tid  = threadIdx.x;
    const int lane = tid & 31;
    const int wave = tid >> 5;
    const int l15  = lane & 15;
    const int half = lane >> 4;        // 0: rows 0-7 / K sel 0,1 ; 1: rows 8-15 / K sel 2,3
    const int dsel = half * 2;

    // ||e||^2 for all 1024 codes (one-time, tiny)
    #pragma unroll
    for (int j = 0; j < 4; ++j) {
        int code = tid + j * 256;
        const v4f* p = reinterpret_cast<const v4f*>(ew + code * EDIM);
        float s = 0.0f;
        #pragma unroll
        for (int q = 0; q < 16; ++q) {
            v4f v = p[q];
            s += v.x * v.x + v.y * v.y + v.z * v.z + v.w * v.w;
        }
        e2[code] = s;
    }

    // A fragments for 2 pixel-tiles. f32 16x4 A layout:
    //   lanes 0-15: m=lane, VGPR0=K0, VGPR1=K1 ; lanes 16-31: m=lane-16, K2/K3.
    v2f  af[2][16];
    float best[2][8];
    int   bidx[2][8];
    int   tbase[2];
    #pragma unroll
    for (int t = 0; t < 2; ++t) {
        int tt = (blockIdx.x * 8 + wave) * 2 + t;      // 0..4095
        int n0 = tt * 16;
        tbase[t] = n0;
        int b  = n0 >> 12;
        int h  = (n0 >> 6) & 63;
        int w0 = n0 & 63;
        const float* zb = z + (size_t)b * 262144 + h * 64 + w0 + l15;
        #pragma unroll
        for (int kf = 0; kf < 16; ++kf) {
            int k = kf * 4 + dsel;
            v2f a;
            a.x = zb[(size_t)k * 4096];
            a.y = zb[(size_t)(k + 1) * 4096];
            af[t][kf] = a;
        }
        #pragma unroll
        for (int i = 0; i < 8; ++i) { best[t][i] = 3.4e38f; bidx[t][i] = 0; }
    }
    __syncthreads();

    #pragma unroll 1
    for (int ch = 0; ch < 8; ++ch) {
        // stage 128 codes (32 KB): coalesced global reads, conflict-free LDS
        #pragma unroll
        for (int it = 0; it < 32; ++it) {
            int e    = it * 256 + tid;          // 0..8191
            int code = e >> 6;
            int k    = e & 63;
            ebuf[code * 66 + k] = ew[ch * 8192 + e];
        }
        __syncthreads();

        #pragma unroll 1
        for (int ct = 0; ct < 8; ++ct) {
            v8f acc[2];
            #pragma unroll
            for (int t = 0; t < 2; ++t)
                acc[t] = (v8f){0.f, 0.f, 0.f, 0.f, 0.f, 0.f, 0.f, 0.f};

            // B 4x16 layout: lanes 0-15 hold rows K0/K1 (n=lane),
            // lanes 16-31 rows K2/K3 -> contiguous (k,k+1) pair in row-major
            // codebook. Each B fragment feeds both pixel-tiles.
            int rb = (ct * 16 + l15) * 66 + dsel;
            #pragma unroll
            for (int kf = 0; kf < 16; ++kf) {
                v2f bf = *reinterpret_cast<const v2f*>(&ebuf[rb + kf * 4]);
                #pragma unroll
                for (int t = 0; t < 2; ++t) {
                    acc[t] = __builtin_amdgcn_wmma_f32_16x16x4_f32(
                        false, af[t][kf], false, bf, (short)0, acc[t],
                        false, false);
                }
            }
            int   n   = ch * 128 + ct * 16 + l15;
            float e2n = e2[n];
            #pragma unroll
            for (int t = 0; t < 2; ++t) {
                #pragma unroll
                for (int i = 0; i < 8; ++i) {
                    float d = fmaf(-2.0f, acc[t][i], e2n);  // ||z||^2 const/row
                    if (d < best[t][i]) { best[t][i] = d; bidx[t][i] = n; }
                }
            }
        }
        __syncthreads();
    }

    // argmin reduce across the 16 lanes of each half (ties -> lower index)
    #pragma unroll
    for (int t = 0; t < 2; ++t) {
        #pragma unroll
        for (int i = 0; i < 8; ++i) {
            float bd = best[t][i];
            int   bi = bidx[t][i];
            #pragma unroll
            for (int off = 8; off >= 1; off >>= 1) {
                float od = __shfl_xor(bd, off, 16);
                int   oi = __shfl_xor(bi, off, 16);
                if (od < bd || (od == bd && oi < bi)) { bd = od; bi = oi; }
            }
            best[t][i] = bd; bidx[t][i] = bi;
        }
        if (l15 < 8) {
            float bd = best[t][0];
            int   bi = bidx[t][0];
            #pragma unroll
            for (int i = 1; i < 8; ++i)
                if (l15 == i) { bd = best[t][i]; bi = bidx[t][i]; }
            int p = tbase[t] + half * 8 + l15;      // row m = half*8 + slot
            out_index[p] = (float)bi;
            atomicAdd(&ws_counts[bi], 1.0f);
        }
    }
}

// ---------------------------------------------------------------------------
// k2: quantize gather (B,C,H,W layout) + vq_loss accumulation
// ---------------------------------------------------------------------------
__global__ __launch_bounds__(256) void vq_quantize_loss_kernel(
    const float* __restrict__ z, const float* __restrict__ ew,
    const float* __restrict__ idxf, float* __restrict__ out_q,
    float* __restrict__ ws_loss)
{
    int gid = blockIdx.x * 256 + threadIdx.x;
    int f = gid * 4;                    // flat (b,c,h,w), 4 consecutive w
    int w = f & 63;
    int h = (f >> 6) & 63;
    int c = (f >> 12) & 63;
    int b = f >> 18;
    int nbase = b * 4096 + h * 64 + w;

    v4f zv = *reinterpret_cast<const v4f*>(z + f);
    v4f qv;
    float ls = 0.0f;
    #pragma unroll
    for (int j = 0; j < 4; ++j) {
        int   idx = (int)(idxf[nbase + j] + 0.5f);
        float q   = ew[idx * EDIM + c];
        qv[j] = q;
        float d = q - zv[j];
        ls += d * d;
    }
    *reinterpret_cast<v4f*>(out_q + f) = qv;

    #pragma unroll
    for (int off = 16; off >= 1; off >>= 1) ls += __shfl_down(ls, off, 32);
    if ((threadIdx.x & 31) == 0) atomicAdd(ws_loss, ls);
}

// ---------------------------------------------------------------------------
// k3: one-hot fill, one row (1024 floats) per block, fully coalesced B128
// ---------------------------------------------------------------------------
__global__ __launch_bounds__(256) void vq_onehot_kernel(
    const float* __restrict__ idxf, float* __restrict__ onehot)
{
    int n   = blockIdx.x;
    int idx = (int)(idxf[n] + 0.5f);
    int c0  = threadIdx.x * 4;
    v4f v;
    v.x = (c0 + 0 == idx) ? 1.0f : 0.0f;
    v.y = (c0 + 1 == idx) ? 1.0f : 0.0f;
    v.z = (c0 + 2 == idx) ? 1.0f : 0.0f;
    v.w = (c0 + 3 == idx) ? 1.0f : 0.0f;
    *reinterpret_cast<v4f*>(onehot + (size_t)n * 1024 + c0) = v;
}

// ---------------------------------------------------------------------------
// k4: scalars (vq_loss, perplexity)
// ---------------------------------------------------------------------------
__global__ __launch_bounds__(256) void vq_finalize_kernel(
    const float* __restrict__ ws_counts, const float* __restrict__ ws_loss,
    float* __restrict__ out_loss, float* __restrict__ out_perp)
{
    __shared__ float red[8];
    float s = 0.0f;
    #pragma unroll
    for (int j = 0; j < 4; ++j) {
        float p = ws_counts[threadIdx.x + j * 256] * (1.0f / 65536.0f);
        s += p * logf(p + 1e-10f);
    }
    #pragma unroll
    for (int off = 16; off >= 1; off >>= 1) s += __shfl_down(s, off, 32);
    if ((threadIdx.x & 31) == 0) red[threadIdx.x >> 5] = s;
    __syncthreads();
    if (threadIdx.x == 0) {
        float tot = 0.0f;
        #pragma unroll
        for (int i = 0; i < 8; ++i) tot += red[i];
        *out_perp = expf(-tot);
        *out_loss = (*ws_loss) * (1.25f / (float)NELEM);
    }
}

// ---------------------------------------------------------------------------
extern "C" void kernel_launch(void* const* d_in, const int* in_sizes, int n_in,
                              void* d_out, int out_size, void* d_ws, size_t ws_size,
                              hipStream_t stream) {
    const float* z  = (const float*)d_in[0];
    const float* ew = (const float*)d_in[1];
    float* out    = (float*)d_out;
    float* q      = out;
    float* loss   = out + OFF_LOSS;
    float* perp   = out + OFF_PERP;
    float* onehot = out + OFF_ONEHOT;
    float* idxf   = out + OFF_INDEX;
    float* ws     = (float*)d_ws;      // [0..1023] counts, [1024] loss sum

    vq_zero_ws_kernel<<<5, 256, 0, stream>>>(ws);
    vq_argmin_kernel<<<256, 256, 0, stream>>>(z, ew, idxf, ws);
    vq_quantize_loss_kernel<<<4096, 256, 0, stream>>>(z, ew, idxf, q, ws + 1024);
    vq_onehot_kernel<<<NPIX, 256, 0, stream>>>(idxf, onehot);
    vq_finalize_kernel<<<1, 256, 0, stream>>>(ws, ws + 1024, loss, perp);
}